// Model_10419590660202
// MI455X (gfx1250) — compile-verified
//
#include <hip/hip_runtime.h>
#include <hip/hip_bf16.h>

#define NU 200000
#define NM 40000
#define NE 2000000
#define HDIM 64

typedef float v2f __attribute__((ext_vector_type(2)));
typedef float v8f __attribute__((ext_vector_type(8)));

// ---------------------------------------------------------------------------
// Per-destination edge counts (degree), as float for later 1/max(cnt,1).
// ---------------------------------------------------------------------------
__global__ __launch_bounds__(256) void count_edges(const int* __restrict__ edst,
                                                   float* __restrict__ cnt, int e_total) {
    int e = blockIdx.x * blockDim.x + threadIdx.x;
    if (e < e_total) atomicAdd(&cnt[edst[e]], 1.0f);
}

// ---------------------------------------------------------------------------
// Scatter-add of source features into destination aggregate.
// One thread per (edge, 4-feature chunk): float4 gather + 4 f32 atomics.
// Tables are L2-resident (51MB + 10MB << 192MB), so atomics resolve on-chip.
// ---------------------------------------------------------------------------
__global__ __launch_bounds__(256) void scatter_feats(const float4* __restrict__ xsrc4,
                                                     const int* __restrict__ esrc,
                                                     const int* __restrict__ edst,
                                                     float* __restrict__ agg,
                                                     long long total /* NE*16 */) {
    long long idx = (long long)blockIdx.x * blockDim.x + threadIdx.x;
    if (idx >= total) return;
    int e  = (int)(idx >> 4);
    int c4 = (int)(idx & 15);                   // which float4 chunk of the row
    int s  = esrc[e];
    int d  = edst[e];
    float4 v = xsrc4[(long long)s * (HDIM / 4) + c4];
    float* dst = &agg[(long long)d * HDIM + c4 * 4];
    atomicAdd(dst + 0, v.x);
    atomicAdd(dst + 1, v.y);
    atomicAdd(dst + 2, v.z);
    atomicAdd(dst + 3, v.w);
}

// ---------------------------------------------------------------------------
// One GEMM pass: acc[4] += (M row-block [16x64] * sc) @ W (from LDS, K-pair
// interleaved layout: sW[(k>>1)*128 + col*2 + (k&1)]).
// SCALE selected at compile time so the X@Wr pass has no scaling VALU.
// ---------------------------------------------------------------------------
template <bool SCALE>
__device__ __forceinline__ void gemm_pass(const float* __restrict__ M,
                                          const float* __restrict__ sW,
                                          float sc, int arow, int half, int l15,
                                          v8f acc[4]) {
    #pragma unroll 4
    for (int kk = 0; kk < HDIM; kk += 4) {
        int ka = kk + half * 2;                  // even -> 8B-aligned b64 load
        v2f afrag = *(const v2f*)(M + (long long)arow * HDIM + ka);
        if (SCALE) { afrag.x *= sc; afrag.y *= sc; }
        const float* wrow = sW + (ka >> 1) * 128;   // K-pair interleaved
        #pragma unroll
        for (int ct = 0; ct < 4; ++ct) {
            v2f bfrag = *(const v2f*)(wrow + (ct * 16 + l15) * 2);
            acc[ct] = __builtin_amdgcn_wmma_f32_16x16x4_f32(
                false, afrag, false, bfrag, (short)0, acc[ct], false, false);
        }
    }
}

// ---------------------------------------------------------------------------
// out = relu?( (A .* 1/max(cnt,1)) @ Wl + X @ Wr + b )
//   A, X : [n,64]  Wl, Wr : [64,64] row-major  b : [64]
// 8 waves / block, each wave computes a 16x64 strip with f32 WMMA (16x16x4).
// Weights staged into LDS (32KB of 320KB WGP LDS), interleaved by K-pair so a
// lane's B fragment is one contiguous ds_load_b64 into an even VGPR pair.
// n must be a multiple of 16 (NU, NM both are).
// ---------------------------------------------------------------------------
__global__ __launch_bounds__(256) void sage_gemm_wmma(const float* __restrict__ A,
                                                      const float* __restrict__ cntA,
                                                      const float* __restrict__ Wl,
                                                      const float* __restrict__ X,
                                                      const float* __restrict__ Wr,
                                                      const float* __restrict__ bias,
                                                      float* __restrict__ out,
                                                      int n, int do_relu) {
    __shared__ float sWl[HDIM * HDIM];
    __shared__ float sWr[HDIM * HDIM];
    __shared__ float sb[HDIM];

    int tid = threadIdx.x;
    // Interleaved store: element W[k][col] -> sW[(k>>1)*128 + col*2 + (k&1)]
    for (int i = tid; i < HDIM * HDIM; i += 256) {
        int k = i >> 6, col = i & 63;
        int di = ((k >> 1) << 7) + (col << 1) + (k & 1);
        sWl[di] = Wl[i];
        sWr[di] = Wr[i];
    }
    if (tid < HDIM) sb[tid] = bias[tid];
    __syncthreads();

    int wave = tid >> 5;
    int lane = tid & 31;
    int row0 = (blockIdx.x * 8 + wave) * 16;
    if (row0 >= n) return;            // wave-uniform: EXEC stays all-ones for WMMA

    int half = lane >> 4;             // 0: K={0,1}, 1: K={2,3} within a k-step
    int l15  = lane & 15;             // A: row m; B/D: column n
    int arow = row0 + l15;            // this lane's A-matrix row (both halves)

    float scaleA = 1.0f / fmaxf(cntA[arow], 1.0f);

    v8f acc[4] = {v8f{}, v8f{}, v8f{}, v8f{}};  // column tiles 0,16,32,48

    gemm_pass<true >(A, sWl, scaleA, arow, half, l15, acc);   // mean(agg) @ Wl
    gemm_pass<false>(X, sWr, 1.0f,   arow, half, l15, acc);   // x_dst     @ Wr

    // D layout: VGPR r -> rows {r (lanes 0-15), r+8 (lanes 16-31)}, col = lane&15
    #pragma unroll
    for (int ct = 0; ct < 4; ++ct) {
        int col = ct * 16 + l15;
        float bv = sb[col];
        #pragma unroll
        for (int r = 0; r < 8; ++r) {
            int m = r + half * 8;
            float v = acc[ct][r] + bv;
            if (do_relu) v = fmaxf(v, 0.0f);
            out[(long long)(row0 + m) * HDIM + col] = v;
        }
    }
}

// ---------------------------------------------------------------------------
// Host-side orchestration (all on `stream`, graph-capture safe).
// ---------------------------------------------------------------------------
extern "C" void kernel_launch(void* const* d_in, const int* in_sizes, int n_in,
                              void* d_out, int out_size, void* d_ws, size_t ws_size,
                              hipStream_t stream) {
    (void)in_sizes; (void)n_in; (void)out_size; (void)ws_size;

    const int*   e_src_um = (const int*)d_in[2];
    const int*   e_dst_um = (const int*)d_in[3];
    const int*   e_src_mu = (const int*)d_in[4];
    const int*   e_dst_mu = (const int*)d_in[5];
    const float* u0       = (const float*)d_in[6];   // user_table (x_user is arange)
    const float* m0       = (const float*)d_in[7];   // movie_table
    const float* Wl1_um   = (const float*)d_in[8];
    const float* Wr1_um   = (const float*)d_in[9];
    const float* Wl1_mu   = (const float*)d_in[10];
    const float* Wr1_mu   = (const float*)d_in[11];
    const float* Wl2_um   = (const float*)d_in[12];
    const float* Wr2_um   = (const float*)d_in[13];
    const float* Wl2_mu   = (const float*)d_in[14];
    const float* Wr2_mu   = (const float*)d_in[15];
    const float* b1_um    = (const float*)d_in[16];
    const float* b1_mu    = (const float*)d_in[17];
    const float* b2_um    = (const float*)d_in[18];
    const float* b2_mu    = (const float*)d_in[19];

    float* u2_out = (float*)d_out;                          // [NU,64]
    float* m2_out = (float*)d_out + (size_t)NU * HDIM;      // [NM,64]

    // Workspace layout (floats)
    float* ws    = (float*)d_ws;
    size_t off   = 0;
    float* agg_u = ws + off; off += (size_t)NU * HDIM;      // 51.2 MB
    float* agg_m = ws + off; off += (size_t)NM * HDIM;      // 10.2 MB
    float* u1    = ws + off; off += (size_t)NU * HDIM;      // 51.2 MB
    float* m1    = ws + off; off += (size_t)NM * HDIM;      // 10.2 MB
    float* cnt_u = ws + off; off += (size_t)NU;
    float* cnt_m = ws + off; off += (size_t)NM;

    const long long scat_total  = (long long)NE * (HDIM / 4);  // 32M (edge,chunk)
    const unsigned  scat_blocks = (unsigned)((scat_total + 255) / 256);
    const unsigned  cnt_blocks  = (NE + 255) / 256;
    const unsigned  gemm_blk_u  = (NU + 127) / 128;            // 8 row-tiles / block
    const unsigned  gemm_blk_m  = (NM + 127) / 128;

    // Degrees (same edge lists for both layers -> compute once)
    hipMemsetAsync(cnt_u, 0, (size_t)NU * sizeof(float), stream);
    hipMemsetAsync(cnt_m, 0, (size_t)NM * sizeof(float), stream);
    count_edges<<<cnt_blocks, 256, 0, stream>>>(e_dst_um, cnt_m, NE);
    count_edges<<<cnt_blocks, 256, 0, stream>>>(e_dst_mu, cnt_u, NE);

    // ---- Layer 1 ----
    hipMemsetAsync(agg_m, 0, (size_t)NM * HDIM * sizeof(float), stream);
    hipMemsetAsync(agg_u, 0, (size_t)NU * HDIM * sizeof(float), stream);
    scatter_feats<<<scat_blocks, 256, 0, stream>>>((const float4*)u0, e_src_um, e_dst_um,
                                                   agg_m, scat_total);
    scatter_feats<<<scat_blocks, 256, 0, stream>>>((const float4*)m0, e_src_mu, e_dst_mu,
                                                   agg_u, scat_total);
    // m1 = relu(mean(agg_m) @ Wl1_um + m0 @ Wr1_um + b1_um)
    sage_gemm_wmma<<<gemm_blk_m, 256, 0, stream>>>(agg_m, cnt_m, Wl1_um, m0, Wr1_um, b1_um,
                                                   m1, NM, 1);
    // u1 = relu(mean(agg_u) @ Wl1_mu + u0 @ Wr1_mu + b1_mu)
    sage_gemm_wmma<<<gemm_blk_u, 256, 0, stream>>>(agg_u, cnt_u, Wl1_mu, u0, Wr1_mu, b1_mu,
                                                   u1, NU, 1);

    // ---- Layer 2 ----
    hipMemsetAsync(agg_m, 0, (size_t)NM * HDIM * sizeof(float), stream);
    hipMemsetAsync(agg_u, 0, (size_t)NU * HDIM * sizeof(float), stream);
    scatter_feats<<<scat_blocks, 256, 0, stream>>>((const float4*)u1, e_src_um, e_dst_um,
                                                   agg_m, scat_total);
    scatter_feats<<<scat_blocks, 256, 0, stream>>>((const float4*)m1, e_src_mu, e_dst_mu,
                                                   agg_u, scat_total);
    // m2 = mean(agg_m) @ Wl2_um + m1 @ Wr2_um + b2_um
    sage_gemm_wmma<<<gemm_blk_m, 256, 0, stream>>>(agg_m, cnt_m, Wl2_um, m1, Wr2_um, b2_um,
                                                   m2_out, NM, 0);
    // u2 = mean(agg_u) @ Wl2_mu + u1 @ Wr2_mu + b2_mu
    sage_gemm_wmma<<<gemm_blk_u, 256, 0, stream>>>(agg_u, cnt_u, Wl2_mu, u1, Wr2_mu, b2_mu,
                                                   u2_out, NU, 0);
}